// GCN_28501402976601
// MI455X (gfx1250) — compile-verified
//
#include <hip/hip_runtime.h>
#include <hip/hip_bf16.h>

typedef __attribute__((ext_vector_type(2))) float v2f;
typedef __attribute__((ext_vector_type(8))) float v8f;

// ---------------------------------------------------------------------------
// Degree / normalization kernels
// ---------------------------------------------------------------------------
__global__ void gcn_deg_init(float* __restrict__ deg, int n) {
    int i = blockIdx.x * blockDim.x + threadIdx.x;
    if (i < n) deg[i] = 1.0f;   // self-loop contributes 1 to every node's degree
}

__global__ void gcn_deg_count(float* __restrict__ deg, const int* __restrict__ dst, int e) {
    int i = blockIdx.x * blockDim.x + threadIdx.x;
    if (i < e) atomicAdd(&deg[dst[i]], 1.0f);
}

__global__ void gcn_dinv(float* __restrict__ deg, int n) {
    int i = blockIdx.x * blockDim.x + threadIdx.x;
    if (i < n) {
        float d = deg[i];
        deg[i] = (d > 0.0f) ? rsqrtf(d) : 0.0f;
    }
}

// ---------------------------------------------------------------------------
// FP32 WMMA GEMM: H[M x N] = X[M x K] * W[K x N]
// One wave computes a 64x16 C block: 4 row-tiles sharing one B fragment, with
// 4 independent v8f accumulators (breaks the WMMA->WMMA RAW chain and gives
// 4x reuse of each B load). K stepped by 4 via V_WMMA_F32_16X16X4_F32.
// ISA lane layouts (CDNA5 ISA §7.12.2):
//   A 16x4 f32 : VGPR0 = {K+0 (lanes 0-15), K+2 (lanes 16-31)}, VGPR1 = {K+1, K+3}
//   B 4x16 f32 : VGPR0 = rows {K+0, K+2}, VGPR1 = rows {K+1, K+3} (N = lane&15)
//   C 16x16 f32: VGPR v = rows {v, v+8}  (lanes 0-15 / 16-31), N = lane&15
// ---------------------------------------------------------------------------
__global__ __launch_bounds__(128)
void gcn_gemm_wmma(const float* __restrict__ X, const float* __restrict__ W,
                   float* __restrict__ H, int M, int K, int N) {
    const int lane    = threadIdx.x & 31;
    const int wave    = (blockIdx.x << 2) + (threadIdx.x >> 5);
    const int tilesN  = N >> 4;
    const int tilesM4 = (M + 63) >> 6;            // groups of 4 row-tiles
    if (wave >= tilesM4 * tilesN) return;

    const int tg   = wave / tilesN;               // 64-row group index
    const int tn   = wave % tilesN;
    const int half = lane >> 4;                   // 0: lanes 0-15, 1: lanes 16-31
    const int l15  = lane & 15;
    const int koff = half * 2;
    const int col  = tn * 16 + l15;

    const float* __restrict__ xr[4];
#pragma unroll
    for (int t = 0; t < 4; ++t) {
        int r = tg * 64 + t * 16 + l15;
        if (r >= M) r = M - 1;                    // clamp for ragged tail group
        xr[t] = X + (size_t)r * K;
    }

    v8f c[4] = {};
    for (int kk = 0; kk < K; kk += 4) {
        v2f b;
        b.x = W[(size_t)(kk + koff)     * N + col];
        b.y = W[(size_t)(kk + koff + 1) * N + col];
        v2f a[4];
#pragma unroll
        for (int t = 0; t < 4; ++t) {
            a[t].x = xr[t][kk + koff];
            a[t].y = xr[t][kk + koff + 1];
        }
#pragma unroll
        for (int t = 0; t < 4; ++t) {
            c[t] = __builtin_amdgcn_wmma_f32_16x16x4_f32(
                       /*neg_a=*/false, a[t], /*neg_b=*/false, b,
                       /*c_mod=*/(short)0, c[t], /*reuse_a=*/false, /*reuse_b=*/false);
        }
    }

#pragma unroll
    for (int t = 0; t < 4; ++t) {
        const int rbase = tg * 64 + t * 16 + half * 8;
#pragma unroll
        for (int v = 0; v < 8; ++v) {
            int r = rbase + v;
            if (r < M) H[(size_t)r * N + col] = c[t][v];
        }
    }
}

// ---------------------------------------------------------------------------
// Self-loop init: A[i,f] = H[i,f] * dinv[i]^2  (float4; also initializes A)
// ---------------------------------------------------------------------------
__global__ void gcn_self_init(const float* __restrict__ H, const float* __restrict__ dinv,
                              float* __restrict__ A, int n, int F) {
    const int chunks = F >> 2;
    long long gid = (long long)blockIdx.x * blockDim.x + threadIdx.x;
    long long tot = (long long)n * chunks;
    if (gid >= tot) return;
    int i = (int)(gid / chunks);
    float dv = dinv[i];
    float s = dv * dv;
    float4 h = *(const float4*)(H + gid * 4);
    float4 o = make_float4(h.x * s, h.y * s, h.z * s, h.w * s);
    *(float4*)(A + gid * 4) = o;
}

// ---------------------------------------------------------------------------
// Edge scatter: A[dst] += H[src] * dinv[src]*dinv[dst]
// One thread per (edge, 4-float chunk): float4 gather + 4x global_atomic_add_f32.
// All of H/A fit in the 192MB L2, so gathers + atomics resolve in L2.
// ---------------------------------------------------------------------------
__global__ void gcn_scatter(const float* __restrict__ H, const float* __restrict__ dinv,
                            const int* __restrict__ src, const int* __restrict__ dst,
                            float* __restrict__ A, int E, int F) {
    const int chunks = F >> 2;
    long long gid = (long long)blockIdx.x * blockDim.x + threadIdx.x;
    long long tot = (long long)E * chunks;
    if (gid >= tot) return;
    int e = (int)(gid / chunks);
    int c = (int)(gid % chunks);
    int s = src[e];
    int d = dst[e];
    float coef = dinv[s] * dinv[d];
    const float4 hv = *(const float4*)(H + (size_t)s * F + (size_t)c * 4);
    float* out = A + (size_t)d * F + (size_t)c * 4;
    atomicAdd(out + 0, hv.x * coef);
    atomicAdd(out + 1, hv.y * coef);
    atomicAdd(out + 2, hv.z * coef);
    atomicAdd(out + 3, hv.w * coef);
}

// ---------------------------------------------------------------------------
// Bias + optional ReLU, in place (float4)
// ---------------------------------------------------------------------------
__global__ void gcn_bias_act(float* __restrict__ A, const float* __restrict__ b,
                             int n, int F, int relu) {
    const int chunks = F >> 2;
    long long gid = (long long)blockIdx.x * blockDim.x + threadIdx.x;
    long long tot = (long long)n * chunks;
    if (gid >= tot) return;
    int c = (int)(gid % chunks);
    float4 bv = *(const float4*)(b + (size_t)c * 4);
    float4 v  = *(const float4*)(A + gid * 4);
    v.x += bv.x; v.y += bv.y; v.z += bv.z; v.w += bv.w;
    if (relu) {
        v.x = fmaxf(v.x, 0.0f); v.y = fmaxf(v.y, 0.0f);
        v.z = fmaxf(v.z, 0.0f); v.w = fmaxf(v.w, 0.0f);
    }
    *(float4*)(A + gid * 4) = v;
}

// ---------------------------------------------------------------------------
// Host-side orchestration
// ---------------------------------------------------------------------------
static inline unsigned cdiv_u(long long a, long long b) {
    return (unsigned)((a + b - 1) / b);
}

extern "C" void kernel_launch(void* const* d_in, const int* in_sizes, int n_in,
                              void* d_out, int out_size, void* d_ws, size_t ws_size,
                              hipStream_t stream) {
    const float* x  = (const float*)d_in[0];
    const int*   ei = (const int*)d_in[1];
    const float* W0 = (const float*)d_in[2];
    const float* b0 = (const float*)d_in[3];
    const float* W1 = (const float*)d_in[4];
    const float* b1 = (const float*)d_in[5];
    const float* W2 = (const float*)d_in[6];
    const float* b2 = (const float*)d_in[7];
    float* out = (float*)d_out;

    const int F_HID = in_sizes[3];               // 128
    const int F_IN  = in_sizes[2] / F_HID;       // 128
    const int F_OUT = in_sizes[7];               // 64
    const int N     = in_sizes[0] / F_IN;        // 50000
    const int E     = in_sizes[1] / 2;           // 800000
    const int* src = ei;
    const int* dst = ei + E;

    // workspace layout: dinv[N] | buf0[N*F_HID] | buf1[N*F_HID]
    float* dinv = (float*)d_ws;
    size_t nAl  = ((size_t)N + 63) & ~(size_t)63;
    float* buf0 = dinv + nAl;
    float* buf1 = buf0 + (size_t)N * F_HID;

    const int T = 256;

    // --- degree + symmetric normalization ---
    gcn_deg_init <<<cdiv_u(N, T), T, 0, stream>>>(dinv, N);
    gcn_deg_count<<<cdiv_u(E, T), T, 0, stream>>>(dinv, dst, E);
    gcn_dinv     <<<cdiv_u(N, T), T, 0, stream>>>(dinv, N);

    // --- layer 0: x(F_IN) -> buf1(F_HID), relu ---
    {
        unsigned tiles = cdiv_u(N, 64) * (F_HID / 16);
        gcn_gemm_wmma<<<cdiv_u(tiles, 4), 128, 0, stream>>>(x, W0, buf0, N, F_IN, F_HID);
        gcn_self_init<<<cdiv_u((long long)N * (F_HID / 4), T), T, 0, stream>>>(buf0, dinv, buf1, N, F_HID);
        gcn_scatter  <<<cdiv_u((long long)E * (F_HID / 4), T), T, 0, stream>>>(buf0, dinv, src, dst, buf1, E, F_HID);
        gcn_bias_act <<<cdiv_u((long long)N * (F_HID / 4), T), T, 0, stream>>>(buf1, b0, N, F_HID, 1);
    }

    // --- layer 1: buf1(F_HID) -> buf1(F_HID), relu ---
    {
        unsigned tiles = cdiv_u(N, 64) * (F_HID / 16);
        gcn_gemm_wmma<<<cdiv_u(tiles, 4), 128, 0, stream>>>(buf1, W1, buf0, N, F_HID, F_HID);
        gcn_self_init<<<cdiv_u((long long)N * (F_HID / 4), T), T, 0, stream>>>(buf0, dinv, buf1, N, F_HID);
        gcn_scatter  <<<cdiv_u((long long)E * (F_HID / 4), T), T, 0, stream>>>(buf0, dinv, src, dst, buf1, E, F_HID);
        gcn_bias_act <<<cdiv_u((long long)N * (F_HID / 4), T), T, 0, stream>>>(buf1, b1, N, F_HID, 1);
    }

    // --- layer 2: buf1(F_HID) -> d_out(F_OUT), no relu ---
    {
        unsigned tiles = cdiv_u(N, 64) * (F_OUT / 16);
        gcn_gemm_wmma<<<cdiv_u(tiles, 4), 128, 0, stream>>>(buf1, W2, buf0, N, F_HID, F_OUT);
        gcn_self_init<<<cdiv_u((long long)N * (F_OUT / 4), T), T, 0, stream>>>(buf0, dinv, out, N, F_OUT);
        gcn_scatter  <<<cdiv_u((long long)E * (F_OUT / 4), T), T, 0, stream>>>(buf0, dinv, src, dst, out, E, F_OUT);
        gcn_bias_act <<<cdiv_u((long long)N * (F_OUT / 4), T), T, 0, stream>>>(out, b2, N, F_OUT, 0);
    }
}